// MaxUnpooling2D_223338299933
// MI455X (gfx1250) — compile-verified
//
#include <hip/hip_runtime.h>

// MaxUnpooling2D: scatter updates[B,H,W,C] into zeroed out[B,2H,2W,C] at
// flat per-batch indices from mask. Pure bandwidth problem (~448 MB moved,
// ~19 us roofline at 23.3 TB/s). No FLOPs -> no WMMA; optimize the CDNA5
// data-movement path: b128 non-temporal loads/stores + global prefetch.

#define B_DIM   16
#define H_IN    64
#define W_IN    64
#define C_DIM   256
#define OH      (H_IN * 2)
#define OW      (W_IN * 2)
#define FLAT_PB ((size_t)OH * OW * C_DIM)                 // 4,194,304 per batch
#define IN_PB   ((size_t)H_IN * W_IN * C_DIM)             // 1,048,576 per batch (2^20)

typedef float  f32x4 __attribute__((ext_vector_type(4)));
typedef int    i32x4 __attribute__((ext_vector_type(4)));

// ---------------------------------------------------------------------------
// Kernel 1: zero the 256 MB output with non-temporal b128 stores.
// One guard per thread, then 4 unconditional b128 NT stores at immediate
// offsets 0/16/32/48 from a single address (64 B per thread, 512 B per wave
// per store instruction, fully coalesced).
// ---------------------------------------------------------------------------
__global__ __launch_bounds__(256) void unpool_zero_fill(f32x4* __restrict__ out,
                                                        long long n_vec4) {
    const size_t tid  = (size_t)blockIdx.x * 256u + threadIdx.x;
    const size_t base = tid * 4;
    if (base + 4 > (size_t)n_vec4) return;   // groups are exact for our sizes

    const f32x4 z = {0.0f, 0.0f, 0.0f, 0.0f};
    f32x4* __restrict__ p = out + base;
    __builtin_nontemporal_store(z, p + 0);   // global_store_b128 th:NT
    __builtin_nontemporal_store(z, p + 1);   // ... offset:16
    __builtin_nontemporal_store(z, p + 2);   // ... offset:32
    __builtin_nontemporal_store(z, p + 3);   // ... offset:48
}

// ---------------------------------------------------------------------------
// Kernel 2: scatter. Each thread handles 4 consecutive input elements
// (int4 mask + float4 updates, NT b128 loads). Mask indices are consecutive
// in C for the pooling-argmax structure -> single aligned b128 NT store in
// the common case; scalar fallback keeps arbitrary masks correct.
// ---------------------------------------------------------------------------
__global__ __launch_bounds__(256) void unpool_scatter(const i32x4* __restrict__ mask,
                                                      const f32x4* __restrict__ upd,
                                                      float* __restrict__ out,
                                                      long long n_elems) {
    const size_t tid  = (size_t)blockIdx.x * 256u + threadIdx.x;
    const size_t elem = tid * 4;
    if (elem >= (size_t)n_elems) return;

    // Prefetch the read streams a few KB ahead (speculative; OOB is dropped).
    __builtin_prefetch((const char*)(mask + tid) + 4096, 0, 0); // global_prefetch_b8
    __builtin_prefetch((const char*)(upd  + tid) + 4096, 0, 0);

    const i32x4 m = __builtin_nontemporal_load(mask + tid);     // global_load_b128 th:NT
    const f32x4 v = __builtin_nontemporal_load(upd  + tid);

    // 4 elements never straddle a batch (IN_PB is a power of two >> 4).
    const size_t b = elem / IN_PB;                               // >> 20
    float* __restrict__ outb = out + b * FLAT_PB;

    if (((m.x & 3) == 0) && (m.y == m.x + 1) && (m.z == m.x + 2) && (m.w == m.x + 3)) {
        __builtin_nontemporal_store(v, (f32x4*)(outb + (size_t)(unsigned)m.x));
    } else {
        __builtin_nontemporal_store(v.x, outb + (size_t)(unsigned)m.x);
        __builtin_nontemporal_store(v.y, outb + (size_t)(unsigned)m.y);
        __builtin_nontemporal_store(v.z, outb + (size_t)(unsigned)m.z);
        __builtin_nontemporal_store(v.w, outb + (size_t)(unsigned)m.w);
    }
}

extern "C" void kernel_launch(void* const* d_in, const int* in_sizes, int n_in,
                              void* d_out, int out_size, void* d_ws, size_t ws_size,
                              hipStream_t stream) {
    (void)n_in; (void)d_ws; (void)ws_size;

    const f32x4* upd  = (const f32x4*)d_in[0];   // updates: float32 [16,64,64,256]
    const i32x4* mask = (const i32x4*)d_in[1];   // mask: int32 flat indices
    float*       out  = (float*)d_out;           // out: float32 [16,128,128,256]

    const long long n_in_elems = (long long)in_sizes[0];     // 16,777,216
    const long long n_out      = (long long)out_size;        // 67,108,864
    const long long n_out_vec4 = n_out / 4;                  // 16,777,216

    // --- Pass 1: zero-fill output (4 x b128 per thread) ---
    {
        const long long threads = (n_out_vec4 + 3) / 4;      // 4,194,304
        const int  block = 256;
        const long long grid = (threads + block - 1) / block; // 16,384
        unpool_zero_fill<<<(unsigned)grid, block, 0, stream>>>((f32x4*)out, n_out_vec4);
    }

    // --- Pass 2: scatter updates per mask (4 elements per thread) ---
    {
        const long long threads = (n_in_elems + 3) / 4;      // 4,194,304
        const int  block = 256;
        const long long grid = (threads + block - 1) / block; // 16,384
        unpool_scatter<<<(unsigned)grid, block, 0, stream>>>(mask, upd, out, n_in_elems);
    }
}